// PagedMistralAttention_11905649344870
// MI455X (gfx1250) — compile-verified
//
#include <hip/hip_runtime.h>
#include <math.h>

typedef __attribute__((ext_vector_type(2))) float v2f;
typedef __attribute__((ext_vector_type(8))) float v8f;

// ---------------------------------------------------------------------------
// Generic tall-skinny GEMM: Y[16][N] = X[16][K] * W[N][K]^T   (fp32, exact)
// One wave32 computes one 16x16 output tile over the full K with
// v_wmma_f32_16x16x4_f32. 4 independent accumulators hide WMMA latency.
// Control flow is wave-uniform -> EXEC stays all-ones (WMMA requirement).
// ---------------------------------------------------------------------------
__global__ __launch_bounds__(128) void gemm16_wmma(
    const float* __restrict__ X,  // [16][K]
    const float* __restrict__ W,  // [N][K]
    float* __restrict__ Y,        // [16][N]
    int N, int K)
{
  const int lane = threadIdx.x & 31;
  const int wave = threadIdx.x >> 5;
  const int tile = blockIdx.x * 4 + wave;        // 16-column tile index
  const int n0   = tile * 16;
  if (n0 >= N) return;                           // wave-uniform

  const int half = lane >> 4;                    // 0: lanes 0-15, 1: 16-31
  const int l16  = lane & 15;

  // A frag: lane=M(l16), regs hold K = k+2*half, k+2*half+1
  const float* arow = X + (size_t)l16 * K + 2 * half;
  // B frag: lane=N(l16), regs hold K rows; B[k][n] = W[n*K+k]
  const float* brow = W + (size_t)(n0 + l16) * K + 2 * half;

  v8f c0 = {}, c1 = {}, c2 = {}, c3 = {};
  for (int k = 0; k < K; k += 16) {
    v2f a0 = *(const v2f*)(arow + k);
    v2f b0 = *(const v2f*)(brow + k);
    v2f a1 = *(const v2f*)(arow + k + 4);
    v2f b1 = *(const v2f*)(brow + k + 4);
    v2f a2 = *(const v2f*)(arow + k + 8);
    v2f b2 = *(const v2f*)(brow + k + 8);
    v2f a3 = *(const v2f*)(arow + k + 12);
    v2f b3 = *(const v2f*)(brow + k + 12);
    c0 = __builtin_amdgcn_wmma_f32_16x16x4_f32(false, a0, false, b0, (short)0, c0, false, false);
    c1 = __builtin_amdgcn_wmma_f32_16x16x4_f32(false, a1, false, b1, (short)0, c1, false, false);
    c2 = __builtin_amdgcn_wmma_f32_16x16x4_f32(false, a2, false, b2, (short)0, c2, false, false);
    c3 = __builtin_amdgcn_wmma_f32_16x16x4_f32(false, a3, false, b3, (short)0, c3, false, false);
  }
  v8f c = (c0 + c1) + (c2 + c3);

  // D layout: VGPR r -> M = r + 8*half, column N = n0 + l16
  const int m = half * 8;
#pragma unroll
  for (int r = 0; r < 8; ++r)
    Y[(size_t)(m + r) * N + (n0 + l16)] = c[r];
}

// ---------------------------------------------------------------------------
// RoPE at single position pos = max(context_lens)-1, applied to q and k.
// q: 16*4096 elements -> qout (workspace); k: 16*1024 -> new_keys (output).
// ---------------------------------------------------------------------------
__global__ __launch_bounds__(256) void rope_kernel(
    const float* __restrict__ qraw, const float* __restrict__ kraw,
    const int* __restrict__ clens,
    float* __restrict__ qout, float* __restrict__ kout)
{
  const int idx = blockIdx.x * 256 + threadIdx.x;   // 0 .. 81919
  int mc = 1;
#pragma unroll
  for (int i = 0; i < 16; ++i) mc = max(mc, clens[i]);
  const float pos = (float)(mc - 1);

  const float* src; float* dst; int i;
  if (idx < 65536) { src = qraw; dst = qout; i = idx; }
  else             { src = kraw; dst = kout; i = idx - 65536; }

  const int d = i & 127;
  const int j = d & 63;
  // inv_freq = 10000^(-j/64) = exp2(-j * log2(10000)/64)
  const float freq = pos * exp2f(-(float)j * 0.2076205059304601f);
  const float c = __cosf(freq), s = __sinf(freq);
  const float x = src[i];
  const float rot = (d < 64) ? -src[i + 64] : src[i - 64];
  dst[i] = x * c + rot * s;
}

// ---------------------------------------------------------------------------
// Paged GQA decode attention. One block per (batch, kv_head); 256 threads.
// BLOCK_SIZE=16 => each 16-position span is exactly one physical KV block.
// Bandwidth-optimal: each needed K/V byte read once, b128/b32 coalesced.
// ---------------------------------------------------------------------------
#define MAXC 2048
__global__ __launch_bounds__(256) void attn_kernel(
    const float* __restrict__ qr,      // [16][32][128] (rope'd)
    const float* __restrict__ kcache,  // [2048][16][8][128]
    const float* __restrict__ vcache,  // [2048][16][8][128]
    const int* __restrict__ btab,      // [16][128]
    const int* __restrict__ clens,     // [16]
    float* __restrict__ attn)          // [16][32][128]
{
  __shared__ float q_s[512];           // 4 groups x 128 dims
  __shared__ float sc[4 * MAXC];       // scores -> probs
  __shared__ int   bt_s[128];
  __shared__ float gsum[4];
  __shared__ float prt[512];

  const int b    = blockIdx.x >> 3;
  const int kv   = blockIdx.x & 7;
  const int t    = threadIdx.x;
  const int lane = t & 31;
  const int wave = t >> 5;
  const int ctx  = clens[b];

  for (int i = t; i < 512; i += 256) q_s[i] = qr[(size_t)b * 4096 + kv * 512 + i];
  if (t < 128) bt_s[t] = btab[b * 128 + t];
  __syncthreads();

  const float scale = 0.08838834764831845f;   // 1/sqrt(128)

  // --- scores: wave w handles positions s = w, w+8, ... ---
  for (int s = wave; s < ctx; s += 8) {
    const int phys = bt_s[s >> 4];
    const float* krow = kcache + ((size_t)phys * 16 + (s & 15)) * 1024 + kv * 128;
    const float4 kf = *(const float4*)(krow + lane * 4);   // 512B/wave, coalesced
    const int di = lane * 4;
    float d0 = q_s[di]*kf.x + q_s[di+1]*kf.y + q_s[di+2]*kf.z + q_s[di+3]*kf.w;
    float d1 = q_s[128+di]*kf.x + q_s[129+di]*kf.y + q_s[130+di]*kf.z + q_s[131+di]*kf.w;
    float d2 = q_s[256+di]*kf.x + q_s[257+di]*kf.y + q_s[258+di]*kf.z + q_s[259+di]*kf.w;
    float d3 = q_s[384+di]*kf.x + q_s[385+di]*kf.y + q_s[386+di]*kf.z + q_s[387+di]*kf.w;
#pragma unroll
    for (int off = 16; off; off >>= 1) {
      d0 += __shfl_xor(d0, off, 32);
      d1 += __shfl_xor(d1, off, 32);
      d2 += __shfl_xor(d2, off, 32);
      d3 += __shfl_xor(d3, off, 32);
    }
    if (lane == 0) {
      sc[s]            = d0 * scale;
      sc[MAXC + s]     = d1 * scale;
      sc[2 * MAXC + s] = d2 * scale;
      sc[3 * MAXC + s] = d3 * scale;
    }
  }
  __syncthreads();

  // --- softmax: waves 0-3 each own one group ---
  if (wave < 4) {
    float m = -3.402823466e38f;
    for (int s = lane; s < ctx; s += 32) m = fmaxf(m, sc[wave * MAXC + s]);
#pragma unroll
    for (int off = 16; off; off >>= 1) m = fmaxf(m, __shfl_xor(m, off, 32));
    float sum = 0.f;
    for (int s = lane; s < ctx; s += 32) {
      const float p = __expf(sc[wave * MAXC + s] - m);
      sc[wave * MAXC + s] = p;
      sum += p;
    }
#pragma unroll
    for (int off = 16; off; off >>= 1) sum += __shfl_xor(sum, off, 32);
    if (lane == 0) gsum[wave] = sum;
  }
  __syncthreads();

  // --- V accumulation: threads = (part 0/1) x 128 dims, coalesced ---
  const int d = t & 127;
  const int p = t >> 7;
  float a0 = 0.f, a1 = 0.f, a2 = 0.f, a3 = 0.f;
  for (int s = p; s < ctx; s += 2) {
    const int phys = bt_s[s >> 4];
    const float v = vcache[((size_t)phys * 16 + (s & 15)) * 1024 + kv * 128 + d];
    a0 += sc[s] * v;
    a1 += sc[MAXC + s] * v;
    a2 += sc[2 * MAXC + s] * v;
    a3 += sc[3 * MAXC + s] * v;
  }
  if (p == 1) { prt[d] = a0; prt[128 + d] = a1; prt[256 + d] = a2; prt[384 + d] = a3; }
  __syncthreads();
  if (p == 0) {
    a0 += prt[d]; a1 += prt[128 + d]; a2 += prt[256 + d]; a3 += prt[384 + d];
    float* o = attn + (size_t)b * 4096 + kv * 512 + d;
    o[0]   = a0 / gsum[0];
    o[128] = a1 / gsum[1];
    o[256] = a2 / gsum[2];
    o[384] = a3 / gsum[3];
  }
}

// ---------------------------------------------------------------------------
// Host launcher
// ---------------------------------------------------------------------------
extern "C" void kernel_launch(void* const* d_in, const int* in_sizes, int n_in,
                              void* d_out, int out_size, void* d_ws, size_t ws_size,
                              hipStream_t stream) {
  const float* hidden = (const float*)d_in[0];  // [16][1][4096]
  const float* kcache = (const float*)d_in[1];  // [2048][16][8][128]
  const float* vcache = (const float*)d_in[2];
  const int*   btab   = (const int*)d_in[3];    // [16][128]
  const int*   clens  = (const int*)d_in[4];    // [16]
  const float* Wq     = (const float*)d_in[6];  // [4096][4096]
  const float* Wk     = (const float*)d_in[7];  // [1024][4096]
  const float* Wv     = (const float*)d_in[8];  // [1024][4096]
  const float* Wo     = (const float*)d_in[9];  // [4096][4096]

  float* out      = (float*)d_out;
  float* out_attn = out;            // 16*4096
  float* out_nk   = out + 65536;    // 16*1024
  float* out_nv   = out + 81920;    // 16*1024

  float* ws    = (float*)d_ws;
  float* qraw  = ws;                // 65536
  float* kraw  = ws + 65536;        // 16384
  float* qrope = ws + 81920;        // 65536
  float* attnw = ws + 147456;       // 65536

  // Projections (WMMA f32). Wv writes new_values directly (no RoPE on v).
  gemm16_wmma<<<dim3(64), dim3(128), 0, stream>>>(hidden, Wq, qraw,   4096, 4096);
  gemm16_wmma<<<dim3(16), dim3(128), 0, stream>>>(hidden, Wk, kraw,   1024, 4096);
  gemm16_wmma<<<dim3(16), dim3(128), 0, stream>>>(hidden, Wv, out_nv, 1024, 4096);

  // RoPE: q -> workspace, k -> new_keys output. 81920 elems.
  rope_kernel<<<dim3(320), dim3(256), 0, stream>>>(qraw, kraw, clens, qrope, out_nk);

  // Paged attention: one block per (batch, kv_head).
  attn_kernel<<<dim3(128), dim3(256), 0, stream>>>(qrope, kcache, vcache, btab, clens, attnw);

  // Output projection.
  gemm16_wmma<<<dim3(64), dim3(128), 0, stream>>>(attnw, Wo, out_attn, 4096, 4096);
}